// EGAT_LSTM_73504070304134
// MI455X (gfx1250) — compile-verified
//
#include <hip/hip_runtime.h>
#include <cstdint>

#define NN 50000
#define NE 800000
#define D  128
#define DE 16

typedef float v2f __attribute__((ext_vector_type(2)));
typedef float v8f __attribute__((ext_vector_type(8)));

// ---------------- ordered-float <-> uint key (monotonic) ----------------
__device__ __forceinline__ unsigned f2key(float x) {
    unsigned b = __float_as_uint(x);
    return (b & 0x80000000u) ? ~b : (b | 0x80000000u);
}
__device__ __forceinline__ float key2f(unsigned k) {
    unsigned b = (k & 0x80000000u) ? (k ^ 0x80000000u) : ~k;
    return __uint_as_float(b);
}

// ---------------- LSTM: W <- lstm^3(W0[layer]) ----------------
// one block per layer, 512 threads (one per gate), W/hs kept in dynamic LDS
__global__ __launch_bounds__(512) void lstm_evolve(
    const float* __restrict__ W0, const float* __restrict__ Wih,
    const float* __restrict__ Whh, const float* __restrict__ bih,
    const float* __restrict__ bhh, float* __restrict__ Wout)
{
    extern __shared__ float smem[];
    float* sW     = smem;             // D*D  (current sequence, overwritten in place with hs)
    float* sh     = smem + D * D;     // D
    float* sgates = sh + D;           // 4*D

    const int tid   = threadIdx.x;    // gate index 0..511
    const int layer = blockIdx.x;
    const float4* wi4 = (const float4*)(Wih + (size_t)layer * 4 * D * D + (size_t)tid * D);
    const float4* wh4 = (const float4*)(Whh + (size_t)layer * 4 * D * D + (size_t)tid * D);
    const float bsum = bih[layer * 4 * D + tid] + bhh[layer * 4 * D + tid];

    for (int idx = tid; idx < D * D; idx += 512) sW[idx] = W0[layer * D * D + idx];
    float c = 0.f;

    for (int pass = 0; pass < 3; ++pass) {
        if (tid < D) sh[tid] = 0.f;
        c = 0.f;
        __syncthreads();
        for (int t = 0; t < D; ++t) {
            const float4* x4 = (const float4*)(sW + t * D);
            const float4* h4 = (const float4*)sh;
            float acc = bsum;
            #pragma unroll 8
            for (int k = 0; k < D / 4; ++k) {
                float4 a = wi4[k], xv = x4[k];
                float4 b = wh4[k], hv = h4[k];
                acc += a.x * xv.x + a.y * xv.y + a.z * xv.z + a.w * xv.w;
                acc += b.x * hv.x + b.y * hv.y + b.z * hv.z + b.w * hv.w;
            }
            sgates[tid] = acc;
            __syncthreads();            // row t fully consumed; safe to overwrite below
            if (tid < D) {
                float ig = 1.f / (1.f + __expf(-sgates[tid]));
                float fg = 1.f / (1.f + __expf(-sgates[D + tid]));
                float gg = tanhf(sgates[2 * D + tid]);
                float og = 1.f / (1.f + __expf(-sgates[3 * D + tid]));
                c = fg * c + ig * gg;
                float h = og * tanhf(c);
                sh[tid] = h;
                sW[t * D + tid] = h;    // in-place: becomes next pass's input sequence
            }
            __syncthreads();
        }
    }
    for (int idx = tid; idx < D * D; idx += 512) Wout[layer * D * D + idx] = sW[idx];
}

// ---------------- ft = nf @ W via V_WMMA_F32_16X16X4_F32 ----------------
// block = 8 waves; wave w owns col tile n0=16w, block owns row tile m0=16*blockIdx
__global__ __launch_bounds__(256) void gemm_ft_wmma(
    const float* __restrict__ nf, const float* __restrict__ Wm, float* __restrict__ ft)
{
    const int lane = threadIdx.x & 31;
    const int l15  = lane & 15;
    const int hi   = lane >> 4;                 // 0 or 1 (lane half)
    const int n0   = (threadIdx.x >> 5) * 16;   // column tile
    const int m0   = blockIdx.x * 16;           // row tile

    // B tile (all K=128 x 16 cols) resident in 64 VGPRs/lane
    v2f Bv[32];
    #pragma unroll
    for (int s = 0; s < 32; ++s) {
        int r = 4 * s + 2 * hi;                 // B 4x16: VGPR0 = rows k / k+2, VGPR1 = k+1 / k+3
        Bv[s][0] = Wm[r * D + n0 + l15];
        Bv[s][1] = Wm[(r + 1) * D + n0 + l15];
    }

    const float2* nf2 = (const float2*)(nf + (size_t)(m0 + l15) * D);
    v8f c = {0.f, 0.f, 0.f, 0.f, 0.f, 0.f, 0.f, 0.f};
    #pragma unroll
    for (int s = 0; s < 32; ++s) {
        float2 av = nf2[2 * s + hi];            // A 16x4: lane<16 -> K0,K1 ; lane>=16 -> K2,K3
        v2f a; a[0] = av.x; a[1] = av.y;
        c = __builtin_amdgcn_wmma_f32_16x16x4_f32(false, a, false, Bv[s],
                                                  (short)0, c, false, false);
    }
    float* o = ft + (size_t)(m0 + hi * 8) * D + n0 + l15;   // C: VGPR v -> M = v + 8*hi
    #pragma unroll
    for (int v = 0; v < 8; ++v) o[(size_t)v * D] = c[v];
}

// ---------------- per-node attention scalars: one wave per node ----------------
__global__ __launch_bounds__(256) void node_scores(
    const float* __restrict__ nf, const float* __restrict__ aw,
    float* __restrict__ s_src, float* __restrict__ s_dst)
{
    int node = blockIdx.x * (blockDim.x >> 5) + (threadIdx.x >> 5);
    int lane = threadIdx.x & 31;
    if (node >= NN) return;
    float4 x  = ((const float4*)(nf + (size_t)node * D))[lane];
    float4 as = ((const float4*)aw)[lane];
    float4 ad = ((const float4*)(aw + D + DE))[lane];
    float accs = x.x * as.x + x.y * as.y + x.z * as.z + x.w * as.w;
    float accd = x.x * ad.x + x.y * ad.y + x.z * ad.z + x.w * ad.w;
    #pragma unroll
    for (int off = 16; off; off >>= 1) {
        accs += __shfl_down(accs, off);
        accd += __shfl_down(accd, off);
    }
    if (lane == 0) { s_src[node] = accs; s_dst[node] = accd; }
}

// ---------------- edge score + segment-max (ordered-uint atomicMax) ----------------
__global__ void edge_score(const int* __restrict__ src, const int* __restrict__ dst,
                           const float* __restrict__ ef, const float* __restrict__ s_src,
                           const float* __restrict__ s_dst, const float* __restrict__ aw,
                           float* __restrict__ score, unsigned* __restrict__ mxu)
{
    int e = blockIdx.x * blockDim.x + threadIdx.x;
    if (e >= NE) return;
    int s = src[e], d = dst[e];
    float acc = s_src[s] + s_dst[d];
    const float4* e4  = (const float4*)(ef + (size_t)e * DE);
    const float4* ae4 = (const float4*)(aw + D);
    #pragma unroll
    for (int k = 0; k < DE / 4; ++k) {
        float4 v = e4[k], a = ae4[k];
        acc += v.x * a.x + v.y * a.y + v.z * a.z + v.w * a.w;
    }
    score[e] = acc;
    atomicMax(mxu + d, f2key(acc));
}

// ---------------- exp + segment-sum ----------------
__global__ void edge_exp(const int* __restrict__ dst, const float* __restrict__ score,
                         const unsigned* __restrict__ mxu, float* __restrict__ ex,
                         float* __restrict__ den)
{
    int e = blockIdx.x * blockDim.x + threadIdx.x;
    if (e >= NE) return;
    int d = dst[e];
    float v = __expf(score[e] - key2f(mxu[d]));
    ex[e] = v;
    __hip_atomic_fetch_add(den + d, v, __ATOMIC_RELAXED, __HIP_MEMORY_SCOPE_AGENT);
}

// ---------------- weighted scatter: one wave per edge, float4 lanes ----------------
__global__ __launch_bounds__(256) void edge_scatter(
    const int* __restrict__ src, const int* __restrict__ dst,
    const float* __restrict__ ft, const float* __restrict__ ex,
    const float* __restrict__ den, float* __restrict__ out)
{
    int e = blockIdx.x * (blockDim.x >> 5) + (threadIdx.x >> 5);
    int lane = threadIdx.x & 31;
    if (e >= NE) return;
    int s = src[e], d = dst[e];
    float dn = den[d];
    dn = (dn == 0.f) ? 1.f : dn;
    float alpha = ex[e] / dn;
    float4 v = ((const float4*)(ft + (size_t)s * D))[lane];
    float* o = out + (size_t)d * D + lane * 4;
    __hip_atomic_fetch_add(o + 0, v.x * alpha, __ATOMIC_RELAXED, __HIP_MEMORY_SCOPE_AGENT);
    __hip_atomic_fetch_add(o + 1, v.y * alpha, __ATOMIC_RELAXED, __HIP_MEMORY_SCOPE_AGENT);
    __hip_atomic_fetch_add(o + 2, v.z * alpha, __ATOMIC_RELAXED, __HIP_MEMORY_SCOPE_AGENT);
    __hip_atomic_fetch_add(o + 3, v.w * alpha, __ATOMIC_RELAXED, __HIP_MEMORY_SCOPE_AGENT);
}

// ---------------- helpers ----------------
__global__ void zero4(float4* p, int n4) {
    int i = blockIdx.x * blockDim.x + threadIdx.x;
    if (i < n4) p[i] = make_float4(0.f, 0.f, 0.f, 0.f);
}
__global__ void leaky(float* p, int n) {
    int i = blockIdx.x * blockDim.x + threadIdx.x;
    if (i < n) { float x = p[i]; p[i] = x > 0.f ? x : 0.01f * x; }
}

extern "C" void kernel_launch(void* const* d_in, const int* in_sizes, int n_in,
                              void* d_out, int out_size, void* d_ws, size_t ws_size,
                              hipStream_t stream)
{
    // only time-step j = T-1 = 2 contributes to the output
    const int*   src = (const int*)d_in[0] + 2 * NE;
    const int*   dst = (const int*)d_in[1] + 2 * NE;
    const float* nf  = (const float*)d_in[2] + (size_t)2 * NN * D;
    const float* ef  = (const float*)d_in[3] + (size_t)2 * NE * DE;
    const float* W0  = (const float*)d_in[4];
    const float* Wih = (const float*)d_in[5];
    const float* Whh = (const float*)d_in[6];
    const float* bih = (const float*)d_in[7];
    const float* bhh = (const float*)d_in[8];
    const float* aw  = (const float*)d_in[9];
    float* out = (float*)d_out;

    char* ws = (char*)d_ws;
    float*    Wl    = (float*)ws;  ws += (size_t)2 * D * D * 4;
    float*    ft    = (float*)ws;  ws += (size_t)NN * D * 4;
    float*    ssrc  = (float*)ws;  ws += (size_t)NN * 4;
    float*    sdst  = (float*)ws;  ws += (size_t)NN * 4;
    float*    score = (float*)ws;  ws += (size_t)NE * 4;
    float*    ex    = (float*)ws;  ws += (size_t)NE * 4;
    unsigned* mxu   = (unsigned*)ws; ws += (size_t)NN * 4;
    float*    den   = (float*)ws;  ws += (size_t)NN * 4;

    // W_i = lstm^3(W0[i]) for both layers (independent -> 2 blocks)
    size_t lstm_smem = (size_t)(D * D + D + 4 * D) * sizeof(float);
    lstm_evolve<<<2, 512, lstm_smem, stream>>>(W0, Wih, Whh, bih, bhh, Wl);

    for (int layer = 0; layer < 2; ++layer) {
        const float* nfi = (layer == 0) ? nf : out;
        const float* awl = aw + layer * (2 * D + DE);

        gemm_ft_wmma<<<NN / 16, 256, 0, stream>>>(nfi, Wl + layer * D * D, ft);
        node_scores<<<(NN + 7) / 8, 256, 0, stream>>>(nfi, awl, ssrc, sdst);
        zero4<<<(NN / 4 + 255) / 256, 256, 0, stream>>>((float4*)mxu, NN / 4);
        zero4<<<(NN / 4 + 255) / 256, 256, 0, stream>>>((float4*)den, NN / 4);
        edge_score<<<(NE + 255) / 256, 256, 0, stream>>>(src, dst, ef, ssrc, sdst, awl, score, mxu);
        edge_exp<<<(NE + 255) / 256, 256, 0, stream>>>(dst, score, mxu, ex, den);
        zero4<<<(NN * D / 4 + 255) / 256, 256, 0, stream>>>((float4*)out, NN * D / 4);
        edge_scatter<<<(NE + 7) / 8, 256, 0, stream>>>(src, dst, ft, ex, den, out);
        leaky<<<(NN * D + 255) / 256, 256, 0, stream>>>(out, NN * D);
    }
}